// LoRALinear_6708738916879
// MI455X (gfx1250) — compile-verified
//
#include <hip/hip_runtime.h>
#include <hip/hip_bf16.h>

// ---------------------------------------------------------------------------
// LoRA linear for MI455X (gfx1250, wave32):
//   out[t,o] = sum_i x[t,i]*W[o,i] + b[o] + sum_r (sum_i x[t,i]*Bp[t,r,i]) * Ap[t,o,r]
// Shapes: tokens T = B*S = 4096, D_IN = D_OUT = 1024, R = 8, fp32 everywhere.
//
// Roofline: ~308 MB of HBM traffic (A/B params 256 MB streamed once) ->
// ~13 us floor at 23.3 TB/s.  Base GEMM is 8.6 GFLOP fp32 -> matrix cores
// (v_wmma_f32_16x16x4_f32, full fp32 precision); LoRA einsums are pure
// streaming (VALU + nontemporal loads, O(1) arithmetic intensity).
//
// Phase 1: inter[t,r] = x[t,:] . Bp[t,r,:]   (streaming, VALU, nontemporal)
// Phase 2: double-buffered WMMA fp32 GEMM x@W^T, fused bias + rank-8 epilogue.
// ---------------------------------------------------------------------------

typedef float v2f __attribute__((ext_vector_type(2)));
typedef float v4f __attribute__((ext_vector_type(4)));
typedef float v8f __attribute__((ext_vector_type(8)));

#define TOKENS 4096
#define DIN    1024
#define DOUT   1024
#define RANK   8

// ---------------------------------------------------------------------------
// Kernel 1: inter = einsum('bsi,bsri->bsr').  One wave32 per token.
// Each lane holds 32 elements of x (8 v4f), reused across all 8 ranks.
// B_params is 128 MB touched once -> nontemporal loads.
// ---------------------------------------------------------------------------
__global__ __launch_bounds__(256) void lora_inter_kernel(
    const float* __restrict__ x,
    const float* __restrict__ Bp,
    float* __restrict__ inter) {
  const int lane = threadIdx.x & 31;
  const int wave = threadIdx.x >> 5;
  const int t = blockIdx.x * 8 + wave;            // token id, 8 waves/block

  const v4f* xp = (const v4f*)(x + (size_t)t * DIN);
  v4f xr[8];
#pragma unroll
  for (int j = 0; j < 8; ++j) xr[j] = xp[j * 32 + lane];

#pragma unroll
  for (int r = 0; r < RANK; ++r) {
    const v4f* bp =
        (const v4f*)(Bp + (size_t)t * (RANK * DIN) + (size_t)r * DIN);
    float acc = 0.f;
#pragma unroll
    for (int j = 0; j < 8; ++j) {
      v4f bv = __builtin_nontemporal_load(&bp[j * 32 + lane]);
      acc += xr[j].x * bv.x + xr[j].y * bv.y + xr[j].z * bv.z + xr[j].w * bv.w;
    }
    // wave32 butterfly reduction
#pragma unroll
    for (int off = 16; off > 0; off >>= 1) acc += __shfl_xor(acc, off, 32);
    if (lane == 0) inter[t * RANK + r] = acc;
  }
}

// ---------------------------------------------------------------------------
// Kernel 2: WMMA fp32 GEMM (x @ W^T) + bias + low-rank adaptation epilogue.
//
// Workgroup = 256 threads = 8 waves, computes a 128(M tokens) x 128(N outs)
// output tile.  Waves arranged 2(M) x 4(N): each wave owns 64x32 = 4x2
// fragments of v_wmma_f32_16x16x4_f32 accumulators.
//
// Double-buffered K pipeline (one barrier per K slab):
//   prefetch slab k+1 (global -> regs) | WMMA-consume slab k (LDS) |
//   store regs -> other LDS buffer | barrier.
// LDS: 2 x (128x36 dwords) x 2 tiles + inter tile = 76 KB (4 WGs/WGP).
// Row stride padded to 36 dwords (144 B): 8-byte fragment reads are
// 8-byte aligned and bank-conflict-free across the 16 rows x 2 K-halves.
//
// Fragment layout (ISA 7.12.2, 32-bit A 16x4): lane&15 = row, VGPR pair holds
// {K=k0, K=k0+1} with k0 = 2*(lane>>4).  B (4x16) loaded with the mirrored
// pattern from the W tile stored as [o][k] (W row-major is already B^T).
// ---------------------------------------------------------------------------
#define TM 128
#define TN 128
#define TK 32
#define NKT (DIN / TK)
#define LSTR 36   // padded dword stride per LDS row

__global__ __launch_bounds__(256) void lora_gemm_kernel(
    const float* __restrict__ x,
    const float* __restrict__ W,
    const float* __restrict__ bias,
    const float* __restrict__ Ap,
    const float* __restrict__ inter,
    float* __restrict__ out) {
  __shared__ float xs[2][TM * LSTR];
  __shared__ float wsm[2][TN * LSTR];
  __shared__ float is[TM * RANK];

  const int tid  = threadIdx.x;
  const int lane = tid & 31;
  const int wv   = tid >> 5;        // wave 0..7
  const int wm   = wv & 1;          // 2 waves in M
  const int wn   = wv >> 1;         // 4 waves in N
  const int t0   = blockIdx.x * TM; // token base
  const int o0   = blockIdx.y * TN; // out-feature base

  const int lrow = tid >> 3;   // 0..31: row within a 32-row group
  const int lf4  = tid & 7;    // v4f column within the 32-wide K slab
  const int r15  = lane & 15;
  const int kh   = (lane >> 4) * 2;   // K half select per WMMA A/B layout

  // ---- prologue: load slab 0 into regs, stage into buffer 0 ----
  v4f px[4], pw[4];
#pragma unroll
  for (int p = 0; p < 4; ++p) {
    const int row = lrow + p * 32;
    px[p] = *(const v4f*)(x + (size_t)(t0 + row) * DIN + lf4 * 4);
    pw[p] = *(const v4f*)(W + (size_t)(o0 + row) * DIN + lf4 * 4);
  }
  // stage the 128x8 inter tile (1024 floats = 256 v4f) once
  ((v4f*)is)[tid] = ((const v4f*)(inter + (size_t)t0 * RANK))[tid];
#pragma unroll
  for (int p = 0; p < 4; ++p) {
    const int row = lrow + p * 32;
    *(v4f*)(&xs[0][row * LSTR + lf4 * 4]) = px[p];
    *(v4f*)(&wsm[0][row * LSTR + lf4 * 4]) = pw[p];
  }

  v8f acc[4][2];
#pragma unroll
  for (int mi = 0; mi < 4; ++mi)
#pragma unroll
    for (int ni = 0; ni < 2; ++ni)
#pragma unroll
      for (int v = 0; v < 8; ++v) acc[mi][ni][v] = 0.f;

  __syncthreads();

  // ---- main pipeline: one barrier per K slab ----
  for (int kt = 0; kt < NKT; ++kt) {
    const int cur = kt & 1;
    const bool more = (kt + 1) < NKT;

    // prefetch next slab into registers (latency overlapped with WMMAs)
    if (more) {
      const int kn = (kt + 1) * TK;
#pragma unroll
      for (int p = 0; p < 4; ++p) {
        const int row = lrow + p * 32;
        px[p] = *(const v4f*)(x + (size_t)(t0 + row) * DIN + kn + lf4 * 4);
        pw[p] = *(const v4f*)(W + (size_t)(o0 + row) * DIN + kn + lf4 * 4);
      }
    }

    // consume current slab from LDS
    const float* xb = xs[cur];
    const float* wb = wsm[cur];
#pragma unroll
    for (int kk = 0; kk < TK; kk += 4) {
      v2f a[4], bfr[2];
#pragma unroll
      for (int mi = 0; mi < 4; ++mi)
        a[mi] = *(const v2f*)(xb + (wm * 64 + mi * 16 + r15) * LSTR + kk + kh);
#pragma unroll
      for (int ni = 0; ni < 2; ++ni)
        bfr[ni] = *(const v2f*)(wb + (wn * 32 + ni * 16 + r15) * LSTR + kk + kh);
#pragma unroll
      for (int mi = 0; mi < 4; ++mi)
#pragma unroll
        for (int ni = 0; ni < 2; ++ni)
          acc[mi][ni] = __builtin_amdgcn_wmma_f32_16x16x4_f32(
              false, a[mi], false, bfr[ni], (short)0, acc[mi][ni],
              false, false);
    }

    // store prefetched slab into the other buffer (no reader conflict:
    // previous readers of that buffer finished before the last barrier)
    if (more) {
#pragma unroll
      for (int p = 0; p < 4; ++p) {
        const int row = lrow + p * 32;
        *(v4f*)(&xs[1 - cur][row * LSTR + lf4 * 4]) = px[p];
        *(v4f*)(&wsm[1 - cur][row * LSTR + lf4 * 4]) = pw[p];
      }
    }
    __syncthreads();
  }

  // ---- epilogue: bias + rank-8 adaptation, fused with the single store ----
  // C/D layout: VGPR v, lanes 0-15 -> M=v, N=lane; lanes 16-31 -> M=v+8.
  const int mhi = (lane >> 4) << 3;
#pragma unroll
  for (int ni = 0; ni < 2; ++ni) {
    const int o = o0 + wn * 32 + ni * 16 + r15;
    const float bv = bias[o];
#pragma unroll
    for (int mi = 0; mi < 4; ++mi) {
#pragma unroll
      for (int v = 0; v < 8; ++v) {
        const int mloc = wm * 64 + mi * 16 + v + mhi;
        const int t = t0 + mloc;
        const v4f* ar = (const v4f*)(Ap + ((size_t)t * DOUT + o) * RANK);
        v4f a0 = __builtin_nontemporal_load(&ar[0]);
        v4f a1 = __builtin_nontemporal_load(&ar[1]);
        const v4f i0 = *(const v4f*)(is + mloc * RANK);
        const v4f i1 = *(const v4f*)(is + mloc * RANK + 4);
        float ad = i0.x * a0.x + i0.y * a0.y + i0.z * a0.z + i0.w * a0.w +
                   i1.x * a1.x + i1.y * a1.y + i1.z * a1.z + i1.w * a1.w;
        out[(size_t)t * DOUT + o] = acc[mi][ni][v] + bv + ad;
      }
    }
  }
}

// ---------------------------------------------------------------------------
// Host-side launcher (graph-capture safe: only kernel launches on `stream`).
// d_in order: x, A_params, B_params, W, b.   d_ws holds inter (4096*8 f32).
// ---------------------------------------------------------------------------
extern "C" void kernel_launch(void* const* d_in, const int* in_sizes, int n_in,
                              void* d_out, int out_size, void* d_ws,
                              size_t ws_size, hipStream_t stream) {
  (void)in_sizes; (void)n_in; (void)out_size; (void)ws_size;
  const float* x    = (const float*)d_in[0];
  const float* Ap   = (const float*)d_in[1];
  const float* Bp   = (const float*)d_in[2];
  const float* W    = (const float*)d_in[3];
  const float* bias = (const float*)d_in[4];
  float* out   = (float*)d_out;
  float* inter = (float*)d_ws;   // 4096*8 floats = 128 KB scratch

  // Phase 1: 4096 tokens, one wave each, 8 waves per block.
  lora_inter_kernel<<<TOKENS / 8, 256, 0, stream>>>(x, Bp, inter);

  // Phase 2: 32 x 8 workgroups of 256 threads.
  dim3 grid(TOKENS / TM, DOUT / TN);
  lora_gemm_kernel<<<grid, 256, 0, stream>>>(x, W, bias, Ap, inter, out);
}